// LengthRegulator_55783035240932
// MI455X (gfx1250) — compile-verified
//
#include <hip/hip_runtime.h>
#include <hip/hip_bf16.h>
#include <stdint.h>

typedef float v4f __attribute__((ext_vector_type(4)));

#ifndef USE_ASYNC_LDS
#define USE_ASYNC_LDS 1
#endif

__device__ __forceinline__ void wait_asynccnt0() {
#if defined(__has_builtin)
#if __has_builtin(__builtin_amdgcn_s_wait_asynccnt)
    __builtin_amdgcn_s_wait_asynccnt(0);
    return;
#endif
#endif
    asm volatile("s_wait_asynccnt 0" ::: "memory");
}

// ---------------------------------------------------------------------------
// Kernel 1: per-batch inclusive scan of duration (wave32 shfl scan).
// Writes csum / excl (int) into workspace and mel_len (as float) into d_out.
// One wave (32 threads) per batch; T <= a few hundred so this is negligible.
// ---------------------------------------------------------------------------
__global__ __launch_bounds__(32) void lr_scan_kernel(
    const int* __restrict__ dur, int* __restrict__ csum, int* __restrict__ excl,
    float* __restrict__ mel_out, int T)
{
    const int b    = blockIdx.x;
    const int lane = threadIdx.x;            // 0..31 (wave32)
    const int base = b * T;

    int carry = 0;
    for (int t0 = 0; t0 < T; t0 += 32) {
        const int t = t0 + lane;
        const int d = (t < T) ? dur[base + t] : 0;
        int v = d;
        // inclusive wave scan (Hillis-Steele via shfl_up, width 32)
        #pragma unroll
        for (int off = 1; off < 32; off <<= 1) {
            int n = __shfl_up(v, off, 32);
            if (lane >= off) v += n;
        }
        const int c = carry + v;
        if (t < T) {
            csum[base + t] = c;
            excl[base + t] = c - d;
        }
        carry = __shfl(c, 31, 32);           // broadcast last lane's inclusive sum
    }
    if (lane == 0) mel_out[b] = (float)carry;   // mel_len = csum[:, -1]
}

// ---------------------------------------------------------------------------
// Kernel 2: frame expansion. Grid = (ceil(max_len/4), B), block = 256 (8 waves).
// Each group of 64 lanes handles one output frame: binary-search the cumsum
// (staged in LDS via CDNA5 async global->LDS copies), then copy one 256-float
// row of x and one row of position_enc with 128-bit non-temporal stores.
// ---------------------------------------------------------------------------
__global__ __launch_bounds__(256) void lr_expand_kernel(
    const float* __restrict__ x, const float* __restrict__ pe,
    const int* __restrict__ csum, const int* __restrict__ excl,
    float* __restrict__ out, float* __restrict__ pos,
    int T, int H, int max_len)
{
    __shared__ int s_csum[1024];
    __shared__ int s_excl[1024];
    __shared__ int s_info[8];                // [0..3] idx per group, [4..7] pos_within

    const int b   = blockIdx.y;
    const int tid = threadIdx.x;

#if USE_ASYNC_LDS
    // CDNA5 async global->LDS copy of the per-batch csum/excl arrays.
    for (int i = tid; i < T; i += 256) {
        const uint64_t gcs = (uint64_t)(uintptr_t)(csum + (size_t)b * T + i);
        const uint64_t gex = (uint64_t)(uintptr_t)(excl + (size_t)b * T + i);
        // low 32 bits of a flat LDS pointer == LDS byte offset (aperture mapping)
        const unsigned lcs = (unsigned)(uintptr_t)&s_csum[i];
        const unsigned lex = (unsigned)(uintptr_t)&s_excl[i];
        asm volatile("global_load_async_to_lds_b32 %0, %1, off"
                     :: "v"(lcs), "v"(gcs) : "memory");
        asm volatile("global_load_async_to_lds_b32 %0, %1, off"
                     :: "v"(lex), "v"(gex) : "memory");
    }
    wait_asynccnt0();
#else
    for (int i = tid; i < T; i += 256) {
        s_csum[i] = csum[(size_t)b * T + i];
        s_excl[i] = excl[(size_t)b * T + i];
    }
#endif
    __syncthreads();

    const int g     = tid >> 6;              // frame group 0..3
    const int lane  = tid & 63;              // 64 lanes per frame
    const int frame = blockIdx.x * 4 + g;
    const int mel   = s_csum[T - 1];
    const bool valid = (frame < max_len) && (frame < mel);
    const bool inrange = frame < max_len;

    if (lane == 0 && inrange) {
        // searchsorted(csum, frame, side='right'): first i with csum[i] > frame
        int lo = 0, hi = T;
        while (lo < hi) {
            const int mid = (lo + hi) >> 1;
            if (s_csum[mid] <= frame) lo = mid + 1; else hi = mid;
        }
        const int idx = (lo < T - 1) ? lo : (T - 1);
        s_info[g]     = idx;
        s_info[4 + g] = valid ? (frame - s_excl[idx]) : 0;
        // warm the gathered x row in cache (gfx1250 global_prefetch)
        __builtin_prefetch(x + ((size_t)b * T + idx) * H, 0, 0);
    }
    __syncthreads();

    if (!inrange) return;

    const int idx = s_info[g];
    const int pw  = s_info[4 + g];

    const v4f* __restrict__ xrow = (const v4f*)(x  + ((size_t)b * T + idx) * H);
    const v4f* __restrict__ prow = (const v4f*)(pe + (size_t)pw * H);
    v4f* __restrict__ orow = (v4f*)(out + ((size_t)b * max_len + frame) * H);
    v4f* __restrict__ qrow = (v4f*)(pos + ((size_t)b * max_len + frame) * H);

    for (int h4 = lane; h4 * 4 < H; h4 += 64) {
        v4f vx = xrow[h4];
        v4f vp = prow[h4];
        if (!valid) { vx = 0; vp = 0; }
        // 262 MB output stream: non-temporal 128-bit stores, keep L2 for x/pe
        __builtin_nontemporal_store(vx, orow + h4);
        __builtin_nontemporal_store(vp, qrow + h4);
    }
}

// ---------------------------------------------------------------------------
extern "C" void kernel_launch(void* const* d_in, const int* in_sizes, int n_in,
                              void* d_out, int out_size, void* d_ws, size_t ws_size,
                              hipStream_t stream) {
    const float* x   = (const float*)d_in[0];
    const float* pe  = (const float*)d_in[1];
    const int*   dur = (const int*)d_in[2];
    // d_in[3] is the max_len scalar on device; all dims are derived from sizes:
    const int H       = (int)((long long)in_sizes[0] / in_sizes[2]);   // 256
    const int max_len = in_sizes[1] / H - 1;                           // MAX_POS == max_len == 2000
    const long long per_b = 2LL * max_len * H + 1;
    const int B = (int)((long long)out_size / per_b);                  // 64
    const int T = in_sizes[2] / B;                                     // 160

    int* csum = (int*)d_ws;
    int* excl = csum + (size_t)B * T;

    float* out = (float*)d_out;
    float* pos = out + (size_t)B * max_len * H;
    float* mel = pos + (size_t)B * max_len * H;

    lr_scan_kernel<<<B, 32, 0, stream>>>(dur, csum, excl, mel, T);

    dim3 grid((max_len + 3) / 4, B);
    lr_expand_kernel<<<grid, 256, 0, stream>>>(x, pe, csum, excl, out, pos, T, H, max_len);
}